// RelGraphConv_20864951124317
// MI455X (gfx1250) — compile-verified
//
#include <hip/hip_runtime.h>

// ---------------------------------------------------------------------------
// RelGraphConv (R-GCN) for MI455X / gfx1250, wave32, f32 WMMA 16x16x4.
//
// Pipeline (all on `stream`):
//   1. k_compose : W[r] = sum_b w_comp[r,b] * weight[b]        (16x64x64, ws)
//   2. k_init / k_count / k_scan / k_fill : bucket edges by relation (ws perm)
//   3. k_selfloop: out = feat @ loop_weight^T + bias           (WMMA f32)
//   4. k_edges   : per relation, 16-edge groups -> WMMA f32 GEMM against W_r,
//                  row-scaled by e_w*norm, global_atomic_add_f32 into out.
//
// ws layout: [0,256KB) composed W; then int cnt[16], cursor[16], offs[17];
//            then perm[N_EDGES] (4MB).  Requires ws_size >= ~4.5MB.
// ---------------------------------------------------------------------------

typedef float v2f __attribute__((ext_vector_type(2)));
typedef float v4f __attribute__((ext_vector_type(4)));
typedef float v8f __attribute__((ext_vector_type(8)));

#define N_NODES   100000
#define N_EDGES   1000000
#define IN_FEAT   64
#define OUT_FEAT  64
#define NUM_RELS  16
#define NUM_BASES 4

#define APAD 68            // LDS row stride (floats) to avoid bank conflicts
#define N_TILES   (N_NODES / 16)      // 6250 exactly
#define SL_BLOCKS ((N_TILES + 7) / 8) // 8 waves per 256-thread block
#define EDGE_BLKX 64                  // blocks per relation row

// ---------------------------- basis composition ----------------------------
__global__ void k_compose(const float* __restrict__ weight,
                          const float* __restrict__ w_comp,
                          float* __restrict__ W) {
  int r = blockIdx.x;
  float a0 = w_comp[r * NUM_BASES + 0];
  float a1 = w_comp[r * NUM_BASES + 1];
  float a2 = w_comp[r * NUM_BASES + 2];
  float a3 = w_comp[r * NUM_BASES + 3];
  for (int j = threadIdx.x; j < IN_FEAT * OUT_FEAT; j += blockDim.x) {
    float v = a0 * weight[0 * 4096 + j] + a1 * weight[1 * 4096 + j] +
              a2 * weight[2 * 4096 + j] + a3 * weight[3 * 4096 + j];
    W[r * 4096 + j] = v;
  }
}

// ------------------------------- bucketing ---------------------------------
__global__ void k_init(int* cnt, int* cur) {
  int t = threadIdx.x;
  if (t < NUM_RELS) { cnt[t] = 0; cur[t] = 0; }
}

__global__ void k_count(const int* __restrict__ et, int* __restrict__ cnt) {
  int e = blockIdx.x * blockDim.x + threadIdx.x;
  if (e < N_EDGES) atomicAdd(&cnt[et[e]], 1);
}

__global__ void k_scan(const int* __restrict__ cnt, int* __restrict__ offs) {
  if (threadIdx.x == 0) {
    int s = 0;
    for (int r = 0; r < NUM_RELS; ++r) { offs[r] = s; s += cnt[r]; }
    offs[NUM_RELS] = s;
  }
}

__global__ void k_fill(const int* __restrict__ et, int* __restrict__ cur,
                       const int* __restrict__ offs, int* __restrict__ perm) {
  int e = blockIdx.x * blockDim.x + threadIdx.x;
  if (e < N_EDGES) {
    int r = et[e];
    int p = atomicAdd(&cur[r], 1);
    perm[offs[r] + p] = e;
  }
}

// --------------------------- self-loop + bias ------------------------------
// out[n, o] = bias[o] + sum_k feat[n,k] * loop_w[o,k]
__global__ void __launch_bounds__(256) k_selfloop(const float* __restrict__ feat,
                                                  const float* __restrict__ loop_w,
                                                  const float* __restrict__ bias,
                                                  float* __restrict__ out) {
  __shared__ float sB[OUT_FEAT * APAD];   // sB[n][k] = loop_w[n][k]
  __shared__ float sA[8][16 * APAD];      // per-wave 16x64 A tile

  for (int j = threadIdx.x; j < 4096; j += 256) {
    int n = j >> 6, k = j & 63;
    sB[n * APAD + k] = loop_w[j];
  }
  __syncthreads();

  const int lane = threadIdx.x & 31;
  const int wid  = threadIdx.x >> 5;
  const int half = lane >> 4;       // K-group select (A/B layout, 16x16x4 f32)
  const int lm   = lane & 15;       // M (A) / N (B,D) index within tile
  const int arow = lane >> 1;       // A staging: 2 lanes per row
  const int acol = (lane & 1) * 32;

  float bcol[4];
#pragma unroll
  for (int nt = 0; nt < 4; ++nt) bcol[nt] = bias[nt * 16 + lm];

  float* myA = sA[wid];

  for (int tile = blockIdx.x * 8 + wid; tile < N_TILES; tile += gridDim.x * 8) {
    // stage A = feat[tile*16 .. tile*16+15][0..63] (coalesced b128 loads)
    const float* frow = feat + (size_t)(tile * 16 + arow) * IN_FEAT;
#pragma unroll
    for (int j = 0; j < 8; ++j)
      *(v4f*)&myA[arow * APAD + acol + 4 * j] = *(const v4f*)&frow[acol + 4 * j];
    __builtin_amdgcn_wave_barrier();

    v8f acc[4];
#pragma unroll
    for (int nt = 0; nt < 4; ++nt) acc[nt] = v8f{};

#pragma unroll
    for (int kb = 0; kb < 16; ++kb) {
      int k = kb * 4 + half * 2;
      v2f a = *(const v2f*)&myA[lm * APAD + k];         // A[m][k], A[m][k+1]
#pragma unroll
      for (int nt = 0; nt < 4; ++nt) {
        v2f b = *(const v2f*)&sB[(nt * 16 + lm) * APAD + k]; // B[k][n], B[k+1][n]
        acc[nt] = __builtin_amdgcn_wmma_f32_16x16x4_f32(
            false, a, false, b, (short)0, acc[nt], false, false);
      }
    }

#pragma unroll
    for (int nt = 0; nt < 4; ++nt)
#pragma unroll
      for (int v = 0; v < 8; ++v) {
        int row = v + half * 8;
        out[(size_t)(tile * 16 + row) * OUT_FEAT + nt * 16 + lm] =
            acc[nt][v] + bcol[nt];
      }
    __builtin_amdgcn_wave_barrier();
  }
}

// --------------------------- edge message GEMM -----------------------------
// For relation r: groups of 16 edges; D = A(16x64 gathered feat) * W_r(64x64),
// row m scaled by e_w*norm, atomically added into out[dst].
__global__ void __launch_bounds__(256) k_edges(const float* __restrict__ feat,
                                               const float* __restrict__ W,
                                               const float* __restrict__ e_w,
                                               const float* __restrict__ nrm,
                                               const int* __restrict__ src,
                                               const int* __restrict__ dst,
                                               const int* __restrict__ perm,
                                               const int* __restrict__ cnt,
                                               const int* __restrict__ offs,
                                               float* __restrict__ out) {
  __shared__ float sB[OUT_FEAT * APAD];   // sB[o][i] = W_r[i][o]  (transposed)
  __shared__ float sA[8][16 * APAD];

  const int r = blockIdx.y;
  const float* Wr = W + (size_t)r * 4096;
  for (int j = threadIdx.x; j < 4096; j += 256) {
    int i = j >> 6, o = j & 63;
    sB[o * APAD + i] = Wr[j];
  }
  __syncthreads();

  const int lane = threadIdx.x & 31;
  const int wid  = threadIdx.x >> 5;
  const int half = lane >> 4;
  const int lm   = lane & 15;
  const int arow = lane >> 1;
  const int acol = (lane & 1) * 32;

  const int count = cnt[r];
  const int base  = offs[r];
  const int ngroups = (count + 15) >> 4;
  float* myA = sA[wid];

  for (int g = blockIdx.x * 8 + wid; g < ngroups; g += gridDim.x * 8) {
    // lanes 0..15 fetch edge metadata for the 16-edge group
    int s = 0, d = -1;
    float sc = 0.0f;
    int idx = g * 16 + lane;
    if (lane < 16 && idx < count) {
      int e = perm[base + idx];
      s  = src[e];
      d  = dst[e];
      sc = e_w[e] * nrm[e];
    }

    // gather A rows: feat[src[m]] -> LDS (invalid rows read feat[0], masked later)
    int srow = __shfl(s, arow, 32);
    const float* frow = feat + (size_t)srow * IN_FEAT;
#pragma unroll
    for (int j = 0; j < 8; ++j)
      *(v4f*)&myA[arow * APAD + acol + 4 * j] = *(const v4f*)&frow[acol + 4 * j];
    __builtin_amdgcn_wave_barrier();

    v8f acc[4];
#pragma unroll
    for (int nt = 0; nt < 4; ++nt) acc[nt] = v8f{};

#pragma unroll
    for (int kb = 0; kb < 16; ++kb) {
      int k = kb * 4 + half * 2;
      v2f a = *(const v2f*)&myA[lm * APAD + k];
#pragma unroll
      for (int nt = 0; nt < 4; ++nt) {
        v2f b = *(const v2f*)&sB[(nt * 16 + lm) * APAD + k];
        acc[nt] = __builtin_amdgcn_wmma_f32_16x16x4_f32(
            false, a, false, b, (short)0, acc[nt], false, false);
      }
    }

    // per-lane row metadata for D rows {v + half*8}
    int   dd[8];
    float ss[8];
#pragma unroll
    for (int v = 0; v < 8; ++v) {
      int row = v + half * 8;
      dd[v] = __shfl(d, row, 32);
      ss[v] = __shfl(sc, row, 32);
    }

    // scatter: out[dst, nt*16+lm] += D[row][col] * (e_w*norm)  (L2-resident)
#pragma unroll
    for (int nt = 0; nt < 4; ++nt)
#pragma unroll
      for (int v = 0; v < 8; ++v)
        if (dd[v] >= 0)
          unsafeAtomicAdd(&out[(size_t)dd[v] * OUT_FEAT + nt * 16 + lm],
                          acc[nt][v] * ss[v]);
    __builtin_amdgcn_wave_barrier();
  }
}

// ------------------------------- launcher ----------------------------------
extern "C" void kernel_launch(void* const* d_in, const int* in_sizes, int n_in,
                              void* d_out, int out_size, void* d_ws, size_t ws_size,
                              hipStream_t stream) {
  (void)in_sizes; (void)n_in; (void)out_size; (void)ws_size;
  const float* feat   = (const float*)d_in[0];
  const float* e_w    = (const float*)d_in[1];
  const float* nrm    = (const float*)d_in[2];
  const float* weight = (const float*)d_in[3];
  const float* w_comp = (const float*)d_in[4];
  const float* h_bias = (const float*)d_in[5];
  const float* loop_w = (const float*)d_in[6];
  const int*   src    = (const int*)d_in[7];
  const int*   dst    = (const int*)d_in[8];
  const int*   ety    = (const int*)d_in[9];
  float* out = (float*)d_out;

  float* Wc  = (float*)d_ws;
  int*   ib  = (int*)((char*)d_ws + (size_t)NUM_RELS * IN_FEAT * OUT_FEAT * sizeof(float));
  int* cnt  = ib;
  int* cur  = ib + 16;
  int* offs = ib + 32;
  int* perm = ib + 64;

  k_compose<<<NUM_RELS, 256, 0, stream>>>(weight, w_comp, Wc);
  k_init<<<1, 32, 0, stream>>>(cnt, cur);
  k_count<<<(N_EDGES + 255) / 256, 256, 0, stream>>>(ety, cnt);
  k_scan<<<1, 1, 0, stream>>>(cnt, offs);
  k_fill<<<(N_EDGES + 255) / 256, 256, 0, stream>>>(ety, cur, offs, perm);
  k_selfloop<<<SL_BLOCKS, 256, 0, stream>>>(feat, loop_w, h_bias, out);
  k_edges<<<dim3(EDGE_BLKX, NUM_RELS), 256, 0, stream>>>(
      feat, Wc, e_w, nrm, src, dst, perm, cnt, offs, out);
}